// STU_47820165873726
// MI455X (gfx1250) — compile-verified
//
#include <hip/hip_runtime.h>
#include <hip/hip_bf16.h>
#include <cstdint>

// STU forward for MI455X (gfx1250), restructured from FFT conv into WMMA GEMMs.
//
//   out[l,j] = sum_{t<=l,k} v[l-t,k] * ( Yp[t,k,j] + (-1)^(l-t) Ym[t,k,j] )
//   Yp_k = x @ M_plus[k],  Ym_k = x @ M_minus[k]
//   A[t,k,j] = Yp + (-1)^t Ym   (drives even output rows l)
//   B[t,k,j] = Yp - (-1)^t Ym   (drives odd  output rows l)
//
// Stage 1: proj kernel  -> A,B in workspace (needs 2*32768*768*4 = 192 MiB ws)
// Stage 2: conv kernel  -> triangular [1024 x 32768] @ [32768 x 768] per parity
//          hot loop unmasked; causal tail masked; Src tiles staged to LDS with
//          gfx1250 async-to-LDS copies (ASYNCcnt + s_wait_asynccnt).
//
// All matrix math uses V_WMMA_F32_16X16X4_F32 (full fp32 fidelity).

#define SEQ 2048
#define DIM 768
#define NE  16
#define RTOT (SEQ * NE)   // 32768 combined (t,k) reduction length
#define LDN  136          // LDS row stride (128 data + 8 pad floats)
#define TBLK 2            // t-steps per staged LDS buffer
#define ROWS (16 * TBLK)  // 32 Src rows per buffer

typedef __attribute__((ext_vector_type(2))) float v2f;
typedef __attribute__((ext_vector_type(4))) float v4f;
typedef __attribute__((ext_vector_type(8))) float v8f;

// GCC-style int4 matching the async-to-LDS builtin's pointee type.
typedef int v4i_g __attribute__((vector_size(16)));
typedef __attribute__((address_space(1))) v4i_g* gptr_v4i;   // global
typedef __attribute__((address_space(3))) v4i_g* lptr_v4i;   // LDS

#if defined(__has_builtin)
#  if __has_builtin(__builtin_amdgcn_global_load_async_to_lds_b128)
#    define HAVE_ASYNC_LDS 1
#  endif
#endif
#ifndef HAVE_ASYNC_LDS
#  define HAVE_ASYNC_LDS 0
#endif

__device__ __forceinline__ void wait_async0() {
#if defined(__has_builtin) && __has_builtin(__builtin_amdgcn_s_wait_asynccnt)
    __builtin_amdgcn_s_wait_asynccnt(0);
#else
    asm volatile("s_wait_asynccnt 0x0" ::: "memory");
#endif
}

// Issue one 16-byte async global->LDS copy (falls back to a sync copy).
__device__ __forceinline__ void copy16_to_lds(const float* g, float* l) {
#if HAVE_ASYNC_LDS
    __builtin_amdgcn_global_load_async_to_lds_b128(
        (gptr_v4i)(uintptr_t)g,
        (lptr_v4i)(unsigned)(uintptr_t)l,
        0, 0);
#else
    *(v4f*)l = *(const v4f*)g;
#endif
}

// ---------------------------------------------------------------------------
// Stage 1: Yp/Ym projections fused into A/B.
// Grid: x = DIM/128 = 6, y = SEQ/64 = 32, z = NE = 16. Block = 256 (8 waves).
// ---------------------------------------------------------------------------
__global__ __launch_bounds__(256) void stu_proj_kernel(
    const float* __restrict__ x,    // [SEQ, DIM]
    const float* __restrict__ Mp,   // [NE, DIM, DIM]
    const float* __restrict__ Mm,   // [NE, DIM, DIM]
    float* __restrict__ Asrc,       // [RTOT, DIM]  (t,k) major
    float* __restrict__ Bsrc)       // [RTOT, DIM]
{
    const int k  = blockIdx.z;
    const int t0 = blockIdx.y * 64;
    const int j0 = blockIdx.x * 128;

    const int wave = threadIdx.x >> 5;
    const int lane = threadIdx.x & 31;
    const int half = lane >> 4;
    const int lrow = lane & 15;
    const int wm   = wave >> 2;
    const int wn   = wave & 3;

    const float* mp = Mp + (size_t)k * DIM * DIM;
    const float* mm = Mm + (size_t)k * DIM * DIM;

    int trow[2], jcol[2];
#pragma unroll
    for (int m = 0; m < 2; ++m) trow[m] = t0 + wm * 32 + m * 16 + lrow;
#pragma unroll
    for (int n = 0; n < 2; ++n) jcol[n] = j0 + wn * 32 + n * 16 + lrow;

    v8f accP[2][2] = {};
    v8f accM[2][2] = {};

    for (int d0 = 0; d0 < DIM; d0 += 4) {
        v2f a[2];
#pragma unroll
        for (int m = 0; m < 2; ++m)
            a[m] = *(const v2f*)(x + (size_t)trow[m] * DIM + d0 + 2 * half);

        v2f bp[2], bm[2];
#pragma unroll
        for (int n = 0; n < 2; ++n) {
            size_t r0 = (size_t)(d0 + 2 * half) * DIM + jcol[n];
            bp[n].x = mp[r0];       bp[n].y = mp[r0 + DIM];
            bm[n].x = mm[r0];       bm[n].y = mm[r0 + DIM];
        }

#pragma unroll
        for (int m = 0; m < 2; ++m)
#pragma unroll
            for (int n = 0; n < 2; ++n) {
                accP[m][n] = __builtin_amdgcn_wmma_f32_16x16x4_f32(
                    false, a[m], false, bp[n], (short)0, accP[m][n], false, false);
                accM[m][n] = __builtin_amdgcn_wmma_f32_16x16x4_f32(
                    false, a[m], false, bm[n], (short)0, accM[m][n], false, false);
            }
    }

#pragma unroll
    for (int m = 0; m < 2; ++m) {
        const int tb = t0 + wm * 32 + m * 16 + half * 8;
#pragma unroll
        for (int n = 0; n < 2; ++n) {
            const int j = j0 + wn * 32 + n * 16 + lrow;
#pragma unroll
            for (int e = 0; e < 8; ++e) {
                const int t   = tb + e;
                const float s = (t & 1) ? -1.0f : 1.0f;
                const float yp = accP[m][n][e];
                const float ym = accM[m][n][e] * s;
                const size_t o = ((size_t)t * NE + k) * DIM + j;
                Asrc[o] = yp + ym;
                Bsrc[o] = yp - ym;
            }
        }
    }
}

// ---------------------------------------------------------------------------
// Stage 2: triangular Toeplitz GEMM.
//   out[2e+p, j] = sum_{t<=2e+p} sum_k v[2e+p-t, k] * Src_p[t*16+k, j]
// Grid: x = DIM/128 = 6, y = (SEQ/2)/64 = 16, z = parity = 2. Block = 256.
// ---------------------------------------------------------------------------
__global__ __launch_bounds__(256) void stu_conv_kernel(
    const float* __restrict__ vfil,  // [SEQ, NE]
    const float* __restrict__ Asrc,  // [RTOT, DIM]
    const float* __restrict__ Bsrc,  // [RTOT, DIM]
    float* __restrict__ out)         // [SEQ, DIM]
{
    __shared__ float lds[2][ROWS * LDN];   // 2 x 17408 B

    const int p  = blockIdx.z;
    const int e0 = blockIdx.y * 64;
    const int j0 = blockIdx.x * 128;
    const float* __restrict__ Src = p ? Bsrc : Asrc;

    const int wave = threadIdx.x >> 5;
    const int lane = threadIdx.x & 31;
    const int half = lane >> 4;
    const int lrow = lane & 15;
    const int wm   = wave >> 2;
    const int wn   = wave & 3;

    // Output row l this lane owns in each M-fragment.
    int lA[2];
#pragma unroll
    for (int m = 0; m < 2; ++m)
        lA[m] = 2 * (e0 + wm * 32 + m * 16 + lrow) + p;

    const int lmin = 2 * e0 + p;           // smallest l in this tile
    const int lmax = lmin + 126;           // largest  l in this tile
    const int nblk      = (lmax + 2) >> 1; // 2-t blocks covering t in [0, lmax]
    const int nblk_safe = (lmin + 1) >> 1; // blocks with 2b+1 <= lmin (no mask)

    v8f acc[2][2] = {};

    // Staging: 32 rows x 128 cols = 1024 float4; 4 per thread, async to LDS.
    const int f0 = threadIdx.x * 4;
    auto stage = [&](int b, int buf) {
#pragma unroll
        for (int q = 0; q < 4; ++q) {
            const int f   = f0 + q;
            const int row = f >> 5;
            const int c4  = (f & 31) << 2;
            copy16_to_lds(Src + ((size_t)b * ROWS + row) * DIM + j0 + c4,
                          &lds[buf][row * LDN + c4]);
        }
    };

    // Per-lane Toeplitz pointers at t = 0 (decrement by NE per t-step).
    const float* wp[2];
#pragma unroll
    for (int m = 0; m < 2; ++m)
        wp[m] = vfil + (size_t)lA[m] * NE + 2 * half;

    stage(0, 0);
#if HAVE_ASYNC_LDS
    wait_async0();
#endif
    __syncthreads();

    int b = 0;
    // ---- Hot loop: fully causal-safe, no masking ----
    for (; b < nblk_safe; ++b) {
        if (b + 1 < nblk) stage(b + 1, (b + 1) & 1);
        const float* lbuf = lds[b & 1];
#pragma unroll
        for (int tt = 0; tt < TBLK; ++tt) {
#pragma unroll
            for (int k0 = 0; k0 < 16; k0 += 4) {
                v2f w[2];
#pragma unroll
                for (int m = 0; m < 2; ++m)
                    w[m] = *(const v2f*)(wp[m] + k0);
                v2f bf[2];
#pragma unroll
                for (int n = 0; n < 2; ++n) {
                    const int j = wn * 32 + n * 16 + lrow;
                    bf[n].x = lbuf[(tt * 16 + k0 + 2 * half)     * LDN + j];
                    bf[n].y = lbuf[(tt * 16 + k0 + 2 * half + 1) * LDN + j];
                }
#pragma unroll
                for (int m = 0; m < 2; ++m)
#pragma unroll
                    for (int n = 0; n < 2; ++n)
                        acc[m][n] = __builtin_amdgcn_wmma_f32_16x16x4_f32(
                            false, w[m], false, bf[n], (short)0, acc[m][n], false, false);
            }
            wp[0] -= NE;
            wp[1] -= NE;
        }
#if HAVE_ASYNC_LDS
        wait_async0();
#endif
        __syncthreads();
    }

    // ---- Causal tail: per-lane masked Toeplitz operand ----
    for (; b < nblk; ++b) {
        if (b + 1 < nblk) stage(b + 1, (b + 1) & 1);
        const float* lbuf = lds[b & 1];
#pragma unroll
        for (int tt = 0; tt < TBLK; ++tt) {
            const int t = 2 * b + tt;
#pragma unroll
            for (int k0 = 0; k0 < 16; k0 += 4) {
                v2f w[2];
#pragma unroll
                for (int m = 0; m < 2; ++m) {
                    const int s  = lA[m] - t;
                    const int sc = s < 0 ? 0 : s;
                    v2f wv = *(const v2f*)(vfil + (size_t)sc * NE + k0 + 2 * half);
                    w[m].x = s < 0 ? 0.0f : wv.x;
                    w[m].y = s < 0 ? 0.0f : wv.y;
                }
                v2f bf[2];
#pragma unroll
                for (int n = 0; n < 2; ++n) {
                    const int j = wn * 32 + n * 16 + lrow;
                    bf[n].x = lbuf[(tt * 16 + k0 + 2 * half)     * LDN + j];
                    bf[n].y = lbuf[(tt * 16 + k0 + 2 * half + 1) * LDN + j];
                }
#pragma unroll
                for (int m = 0; m < 2; ++m)
#pragma unroll
                    for (int n = 0; n < 2; ++n)
                        acc[m][n] = __builtin_amdgcn_wmma_f32_16x16x4_f32(
                            false, w[m], false, bf[n], (short)0, acc[m][n], false, false);
            }
        }
#if HAVE_ASYNC_LDS
        wait_async0();
#endif
        __syncthreads();
    }

    // C/D layout: VGPR q holds M = frag_base + q + half*8, N = lrow.
#pragma unroll
    for (int m = 0; m < 2; ++m) {
        const int eb = e0 + wm * 32 + m * 16 + half * 8;
#pragma unroll
        for (int n = 0; n < 2; ++n) {
            const int j = j0 + wn * 32 + n * 16 + lrow;
#pragma unroll
            for (int q = 0; q < 8; ++q) {
                const int l = 2 * (eb + q) + p;
                out[(size_t)l * DIM + j] = acc[m][n][q];
            }
        }
    }
}

// ---------------------------------------------------------------------------
extern "C" void kernel_launch(void* const* d_in, const int* in_sizes, int n_in,
                              void* d_out, int out_size, void* d_ws, size_t ws_size,
                              hipStream_t stream) {
    (void)in_sizes; (void)n_in; (void)out_size; (void)ws_size;

    const float* x  = (const float*)d_in[0];   // [1, 2048, 768]
    const float* vf = (const float*)d_in[1];   // [2048, 16]
    const float* Mp = (const float*)d_in[2];   // [16, 768, 768]
    const float* Mm = (const float*)d_in[3];   // [16, 768, 768]
    float* out = (float*)d_out;                // [1, 2048, 768]

    float* Asrc = (float*)d_ws;                      // 96 MiB
    float* Bsrc = Asrc + (size_t)RTOT * DIM;         // 96 MiB (needs 192 MiB ws)

    dim3 blk(256);
    dim3 g1(DIM / 128, SEQ / 64, NE);
    hipLaunchKernelGGL(stu_proj_kernel, g1, blk, 0, stream, x, Mp, Mm, Asrc, Bsrc);

    dim3 g2(DIM / 128, (SEQ / 2) / 64, 2);
    hipLaunchKernelGGL(stu_conv_kernel, g2, blk, 0, stream, vf, Asrc, Bsrc, out);
}